// IDPPGN_39290360824125
// MI455X (gfx1250) — compile-verified
//
#include <hip/hip_runtime.h>
#include <hip/hip_bf16.h>
#include <math.h>

// ---------------------------------------------------------------------------
// ID-PPGN forward for MI455X (gfx1250), fp32 end-to-end.
// Heavy GEMMs (1x1 convs + batched 40x40 matmuls) run on V_WMMA_F32_16X16X4_F32.
//
// Assumed d_in order = setup_inputs() dict insertion order, params flattened
// DFS in insertion order (w before b inside each linear):
//   0 x(16,40,64) 1 adj(16,40,40) 2 subgs(128,2) i32 3 num_subg 4 num_node
//   5 inmlp0.w 6 inmlp0.b 7 inmlp1.w 8 inmlp1.b 9 ln.g 10 ln.b 11 idemb(4,64)
//   12+10*blk: mlp1w0,mlp1b0,mlp1w1,mlp1b1,mlp2w0,mlp2b0,mlp2w1,mlp2b1,skw,skb
//   32 set1.w 33 set1.b 34 set2.w 35 set2.b 36 out0.w 37 out0.b 38 out1.w 39 out1.b
// ---------------------------------------------------------------------------

typedef __attribute__((ext_vector_type(2))) float v2f;
typedef __attribute__((ext_vector_type(8))) float v8f;

#define NBATCH 16
#define NMAX   40
#define NHID   64
#define NN     256           // S*PERMDIM = 16*8*2
#define POS    1600          // 40*40
#define PTILES 100           // 1600/16

__device__ __forceinline__ v8f wmma_f32_4(v2f a, v2f b, v8f c) {
  // D = A(16x4 f32) * B(4x16 f32) + C(16x16 f32)
  return __builtin_amdgcn_wmma_f32_16x16x4_f32(false, a, false, b, (short)0, c,
                                               false, false);
}

__device__ __forceinline__ float seluf(float x) {
  const float sc = 1.0507009873554805f, al = 1.6732632423543772f;
  return x > 0.f ? sc * x : sc * al * (expf(x) - 1.f);
}

// ---------------- input MLP + LayerNorm + relu ------------------------------
__global__ void mlp_ln_kernel(const float* __restrict__ x,
                              const float* __restrict__ w0, const float* __restrict__ b0,
                              const float* __restrict__ w1, const float* __restrict__ b1,
                              const float* __restrict__ g,  const float* __restrict__ bb,
                              float* __restrict__ H) {
  __shared__ float xr[NHID], h0[NHID], h1[NHID];
  int row = blockIdx.x, o = threadIdx.x;
  xr[o] = x[(size_t)row * NHID + o];
  __syncthreads();
  float a = b0[o];
  for (int c = 0; c < NHID; ++c) a += xr[c] * w0[c * NHID + o];
  h0[o] = fmaxf(a, 0.f);
  __syncthreads();
  float h = b1[o];
  for (int c = 0; c < NHID; ++c) h += h0[c] * w1[c * NHID + o];
  h1[o] = h;
  __syncthreads();
  float mu = 0.f;
  for (int c = 0; c < NHID; ++c) mu += h1[c];
  mu *= (1.f / NHID);
  float var = 0.f;
  for (int c = 0; c < NHID; ++c) { float d = h1[c] - mu; var += d * d; }
  var *= (1.f / NHID);
  float val = (h - mu) * rsqrtf(var + 1e-5f) * g[o] + bb[o];
  H[(size_t)row * NHID + o] = fmaxf(val, 0.f);
}

// ---------------- build t(N,C,40,40) ----------------------------------------
__global__ void build_T_kernel(const float* __restrict__ adj,
                               const float* __restrict__ H,
                               const int*   __restrict__ subgs,
                               const float* __restrict__ idemb,
                               float* __restrict__ T) {
  size_t idx = (size_t)blockIdx.x * blockDim.x + threadIdx.x;
  const size_t TOT = (size_t)NN * NHID * POS;
  if (idx >= TOT) return;
  int n  = (int)(idx / ((size_t)NHID * POS));
  int rm = (int)(idx % ((size_t)NHID * POS));
  int c  = rm / POS;
  int pq = rm % POS;
  int i  = pq / NMAX, j = pq % NMAX;
  int s  = n >> 1, pm = n & 1, b = s >> 3;   // NSUB=8
  float val;
  if (i != j) {
    val = adj[(size_t)b * POS + j * NMAX + i];   // symmetric anyway
  } else {
    float base = H[((size_t)b * NMAX + i) * NHID + c];
    int n0 = subgs[s * 2 + 0], n1 = subgs[s * 2 + 1];
    int kk = (i == n0) ? 0 : (i == n1) ? 1 : -1;
    if (kk >= 0) {
      int pid = kk ^ pm;                          // ALLPERM[kk][pm] for K=2
      val = (base + 10.f * idemb[pid * NHID + c]) * (float)((kk + 1) * 20);
    } else val = base;
  }
  T[idx] = val;
}

// ---------------- 1x1 conv as WMMA GEMM -------------------------------------
// out[n,o,p] = act( sum_c W[c,o] * in[n,c,p] + bias[o] )
// channels 0..63 from in1, 64..127 from in2 (skip-conv concat).
// Block = 128 threads (4 waves). Wave w owns p-tile (blk%25)*4+w and computes
// ALL 64 output channels (4 accumulators): streaming operand loaded once per
// K-step, weights staged once per block into LDS (stride-80 anti-conflict).
// In-place safe: each wave exclusively owns its 16 positions for all o.
template <int CIN, bool RELU>
__global__ void __launch_bounds__(128)
conv1x1_wmma(const float* __restrict__ in1, const float* __restrict__ in2,
             const float* __restrict__ W, const float* __restrict__ bias,
             float* __restrict__ out) {
  __shared__ float Wl[CIN * 80];
  for (int idx = threadIdx.x; idx < CIN * 64; idx += 128) {
    int r = idx >> 6, c = idx & 63;
    Wl[r * 80 + c] = W[idx];
  }
  __syncthreads();

  int lane = threadIdx.x & 31;
  int wave = threadIdx.x >> 5;
  int m = lane & 15, khi = lane >> 4;
  int n  = blockIdx.x / 25;
  int p0 = ((blockIdx.x % 25) * 4 + wave) * 16;

  v8f acc[4];
#pragma unroll
  for (int ot = 0; ot < 4; ++ot) acc[ot] = (v8f){0.f,0.f,0.f,0.f,0.f,0.f,0.f,0.f};

#pragma unroll
  for (int kb = 0; kb < CIN; kb += 4) {
    int ka = kb + 2 * khi;                // even -> (ka,ka+1) never straddle 64
    const float* src = (ka < NHID) ? in1 : in2;
    int kc = (ka < NHID) ? ka : (ka - NHID);
    v2f a;
    size_t xb = ((size_t)n * NHID + kc) * POS + p0 + m;
    a.x = src[xb];                        // A[m=p, k] = X[c, p0+m]
    a.y = src[xb + POS];
#pragma unroll
    for (int ot = 0; ot < 4; ++ot) {
      v2f b;
      b.x = Wl[ka * 80 + ot * 16 + m];    // B[k, j=o] = W[c, o]
      b.y = Wl[(ka + 1) * 80 + ot * 16 + m];
      acc[ot] = wmma_f32_4(a, b, acc[ot]);
    }
  }

#pragma unroll
  for (int ot = 0; ot < 4; ++ot) {
    int o = ot * 16 + m;
    float bv = bias[o];
    float r[8];
#pragma unroll
    for (int v = 0; v < 8; ++v) {
      float t = acc[ot][v] + bv;
      r[v] = RELU ? fmaxf(t, 0.f) : t;
    }
    // D row p = p0 + 8*khi + v -> 8 consecutive floats per lane
    size_t ob = ((size_t)n * NHID + o) * POS + p0 + 8 * khi;
    *(float4*)(out + ob)     = make_float4(r[0], r[1], r[2], r[3]);
    *(float4*)(out + ob + 4) = make_float4(r[4], r[5], r[6], r[7]);
  }
}

// ---------------- batched 40x40x40 matmul (mult) ----------------------------
// grid = NN*NHID blocks, 288 threads = 9 waves, one 16x16 tile of the
// zero-padded 48x48 product each. Both slices staged in LDS (zero-padded so
// the K-loop is branch-free). In-place into A (global reads all precede the
// staging barrier; no other block touches this batch).
__global__ void __launch_bounds__(288)
bmm40_wmma(const float* __restrict__ A, const float* __restrict__ Bm,
           float* __restrict__ out) {
  __shared__ float As[48 * 40];   // row-major, rows 40..47 zero
  __shared__ float Bs[40 * 48];   // k-major,   cols 40..47 zero
  size_t base = (size_t)blockIdx.x * POS;
  for (int idx = threadIdx.x; idx < 48 * 40; idx += 288)
    As[idx] = (idx < POS) ? A[base + idx] : 0.f;
  for (int idx = threadIdx.x; idx < 40 * 48; idx += 288) {
    int k = idx / 48, j = idx % 48;
    Bs[idx] = (j < NMAX) ? Bm[base + k * NMAX + j] : 0.f;
  }
  __syncthreads();

  int lane = threadIdx.x & 31;
  int wave = threadIdx.x >> 5;            // 0..8
  int ti = wave / 3, tj = wave % 3;
  int m = lane & 15, khi = lane >> 4;
  int i = ti * 16 + m;                    // A row for this lane (may be >=40)
  int j = tj * 16 + m;                    // B col for this lane (may be >=40)

  v8f acc = {0.f, 0.f, 0.f, 0.f, 0.f, 0.f, 0.f, 0.f};
#pragma unroll
  for (int kb = 0; kb < NMAX; kb += 4) {
    int k = kb + 2 * khi;
    v2f a, b;
    a.x = As[i * NMAX + k];               // contiguous pair -> ds_load_b64
    a.y = As[i * NMAX + k + 1];
    b.x = Bs[k * 48 + j];
    b.y = Bs[(k + 1) * 48 + j];
    acc = wmma_f32_4(a, b, acc);
  }

  bool jv = (j < NMAX);
#pragma unroll
  for (int v = 0; v < 8; ++v) {
    int ii = ti * 16 + v + 8 * khi;
    if (ii < NMAX && jv) out[base + (size_t)ii * NMAX + j] = acc[v];
  }
}

// ---------------- diag / offdiag pooling over perms+subgraphs ---------------
// Yb[b*40+i, z] : z<64 -> sum_n diag ; z>=64 -> sum_n (rowsum - diag)
__global__ void reduce_diag_off(const float* __restrict__ T, float* __restrict__ Yb) {
  int b = blockIdx.x / NMAX, i = blockIdx.x % NMAX, z = threadIdx.x;
  float acc = 0.f;
  if (z < NHID) {
    int c = z;
    for (int q = 0; q < 16; ++q) {
      int n = b * 16 + q;
      acc += T[((size_t)n * NHID + c) * POS + (size_t)i * NMAX + i];
    }
  } else {
    int c = z - NHID;
    for (int q = 0; q < 16; ++q) {
      int n = b * 16 + q;
      size_t rb = ((size_t)n * NHID + c) * POS + (size_t)i * NMAX;
      float s = 0.f;
      for (int j = 0; j < NMAX; ++j) s += T[rb + j];
      acc += s - T[rb + i];
    }
  }
  Yb[(size_t)blockIdx.x * 128 + z] = acc;
}

// ---------------- set1 + selu ----------------------------------------------
__global__ void set1_kernel(const float* __restrict__ Yb,
                            const float* __restrict__ w, const float* __restrict__ b,
                            float* __restrict__ Y1) {
  __shared__ float s[128];
  int row = blockIdx.x, z = threadIdx.x;
  s[z] = Yb[(size_t)row * 128 + z];
  __syncthreads();
  float v = b[z];
  for (int k = 0; k < 128; ++k) v += s[k] * w[k * 128 + z];
  Y1[(size_t)row * 128 + z] = seluf(v);
}

// ---------------- node-sum -> set2(selu) -> out0(relu) -> out1 --------------
__global__ void head_kernel(const float* __restrict__ Y1,
                            const float* __restrict__ s2w, const float* __restrict__ s2b,
                            const float* __restrict__ o0w, const float* __restrict__ o0b,
                            const float* __restrict__ o1w, const float* __restrict__ o1b,
                            float* __restrict__ out) {
  __shared__ float s0[128], s1[128];
  int b = blockIdx.x, z = threadIdx.x;
  float acc = 0.f;
  for (int r = 0; r < NMAX; ++r) acc += Y1[((size_t)b * NMAX + r) * 128 + z];
  s0[z] = acc;
  __syncthreads();
  float v = s2b[z];
  for (int k = 0; k < 128; ++k) v += s0[k] * s2w[k * 128 + z];
  s1[z] = seluf(v);
  __syncthreads();
  float u = o0b[z];
  for (int k = 0; k < 128; ++k) u += s1[k] * o0w[k * 128 + z];
  s0[z] = fmaxf(u, 0.f);
  __syncthreads();
  if (z < 10) {
    float o = o1b[z];
    for (int k = 0; k < 128; ++k) o += s0[k] * o1w[k * 10 + z];
    out[b * 10 + z] = o;
  }
}

// ---------------------------------------------------------------------------
extern "C" void kernel_launch(void* const* d_in, const int* in_sizes, int n_in,
                              void* d_out, int out_size, void* d_ws, size_t ws_size,
                              hipStream_t stream) {
  (void)in_sizes; (void)n_in; (void)out_size; (void)ws_size;
  const float* x     = (const float*)d_in[0];
  const float* adj   = (const float*)d_in[1];
  const int*   subgs = (const int*)  d_in[2];
  const float* w_in0 = (const float*)d_in[5];
  const float* b_in0 = (const float*)d_in[6];
  const float* w_in1 = (const float*)d_in[7];
  const float* b_in1 = (const float*)d_in[8];
  const float* ln_g  = (const float*)d_in[9];
  const float* ln_b  = (const float*)d_in[10];
  const float* idemb = (const float*)d_in[11];
  const float* set1w = (const float*)d_in[32];
  const float* set1b = (const float*)d_in[33];
  const float* set2w = (const float*)d_in[34];
  const float* set2b = (const float*)d_in[35];
  const float* o0w   = (const float*)d_in[36];
  const float* o0b   = (const float*)d_in[37];
  const float* o1w   = (const float*)d_in[38];
  const float* o1b   = (const float*)d_in[39];
  float* out = (float*)d_out;

  const size_t BIG = (size_t)NN * NHID * POS;   // 26,214,400 floats
  float* T  = (float*)d_ws;
  float* M1 = T + BIG;
  float* M2 = M1 + BIG;
  float* H  = M2 + BIG;                         // 640*64
  float* Yb = H + (size_t)NBATCH * NMAX * NHID; // 640*128
  float* Y1 = Yb + (size_t)NBATCH * NMAX * 128; // 640*128

  mlp_ln_kernel<<<NBATCH * NMAX, NHID, 0, stream>>>(x, w_in0, b_in0, w_in1, b_in1,
                                                    ln_g, ln_b, H);
  {
    size_t TOT = BIG;
    int blocks = (int)((TOT + 255) / 256);
    build_T_kernel<<<blocks, 256, 0, stream>>>(adj, H, subgs, idemb, T);
  }

  const int CONV_GRID = NN * 25;                // 6400 blocks, 4 p-tiles each
  for (int blk = 0; blk < 2; ++blk) {
    const float* m1w0 = (const float*)d_in[12 + blk * 10 + 0];
    const float* m1b0 = (const float*)d_in[12 + blk * 10 + 1];
    const float* m1w1 = (const float*)d_in[12 + blk * 10 + 2];
    const float* m1b1 = (const float*)d_in[12 + blk * 10 + 3];
    const float* m2w0 = (const float*)d_in[12 + blk * 10 + 4];
    const float* m2b0 = (const float*)d_in[12 + blk * 10 + 5];
    const float* m2w1 = (const float*)d_in[12 + blk * 10 + 6];
    const float* m2b1 = (const float*)d_in[12 + blk * 10 + 7];
    const float* skw  = (const float*)d_in[12 + blk * 10 + 8];
    const float* skb  = (const float*)d_in[12 + blk * 10 + 9];

    conv1x1_wmma<64,  true ><<<CONV_GRID, 128, 0, stream>>>(T,  nullptr, m1w0, m1b0, M1);
    conv1x1_wmma<64,  true ><<<CONV_GRID, 128, 0, stream>>>(M1, nullptr, m1w1, m1b1, M1);
    conv1x1_wmma<64,  true ><<<CONV_GRID, 128, 0, stream>>>(T,  nullptr, m2w0, m2b0, M2);
    conv1x1_wmma<64,  true ><<<CONV_GRID, 128, 0, stream>>>(M2, nullptr, m2w1, m2b1, M2);
    bmm40_wmma<<<NN * NHID, 288, 0, stream>>>(M1, M2, M1);          // mult -> M1
    conv1x1_wmma<128, false><<<CONV_GRID, 128, 0, stream>>>(T, M1, skw, skb, T);
  }

  reduce_diag_off<<<NBATCH * NMAX, 128, 0, stream>>>(T, Yb);
  set1_kernel<<<NBATCH * NMAX, 128, 0, stream>>>(Yb, set1w, set1b, Y1);
  head_kernel<<<NBATCH, 128, 0, stream>>>(Y1, set2w, set2b, o0w, o0b, o1w, o1b, out);
}